// ContextualAttention_Enhance_8589934592442
// MI455X (gfx1250) — compile-verified
//
#include <hip/hip_runtime.h>
#include <hip/hip_bf16.h>
#include <math.h>
#include <float.h>

typedef float v2f __attribute__((ext_vector_type(2)));
typedef float v8f __attribute__((ext_vector_type(8)));

#define NB   4
#define Cc   64
#define CIc  16
#define Hh   96
#define Wc   96
#define HW   (Hh*Wc)     // 9216
#define PSc  7
#define WSc  9
#define Kk   32
#define WHc  4
#define PPc  3
#define nHc  24
#define NQc  576
#define HCc  102         // H + 2*PP

__device__ __forceinline__ int reflect_idx(int r) {
  if (r < 0) r = -r;
  if (r >= Hh) r = 2*Hh - 2 - r;
  return r;
}
__device__ __forceinline__ int clampi(int v, int lo, int hi) {
  return v < lo ? lo : (v > hi ? hi : v);
}

// ---------------------------------------------------------------------------
// Kernel 1: fused theta/phi/g 1x1 conv.  Per wave: one 16(CI)x16(pixel) tile,
// K=64 via 16 steps of v_wmma_f32_16x16x4_f32, B (vid chunk) loaded once and
// reused by all three weight matrices.
// ---------------------------------------------------------------------------
__global__ __launch_bounds__(256) void proj_kernel(
    const float* __restrict__ vid,
    const float* __restrict__ tw, const float* __restrict__ tb,
    const float* __restrict__ pw, const float* __restrict__ pb,
    const float* __restrict__ gw, const float* __restrict__ gb,
    float* __restrict__ thetaO, float* __restrict__ phiO, float* __restrict__ gO)
{
  int wave = threadIdx.x >> 5;
  int lane = threadIdx.x & 31;
  int gt   = blockIdx.x * 8 + wave;       // 0 .. NB*576-1
  int n    = gt / (HW/16);
  int tile = gt % (HW/16);
  int col  = lane & 15;
  int hf   = lane >> 4;
  int pix  = tile*16 + col;
  int m    = col;                          // A-matrix row (M = lane%16)

  v8f ct = {0.f,0.f,0.f,0.f,0.f,0.f,0.f,0.f};
  v8f cp = {0.f,0.f,0.f,0.f,0.f,0.f,0.f,0.f};
  v8f cg = {0.f,0.f,0.f,0.f,0.f,0.f,0.f,0.f};

  const float* X = vid + (size_t)n * Cc * HW;
  #pragma unroll
  for (int kk = 0; kk < 16; ++kk) {
    int kb = kk*4 + 2*hf;                  // K index for v0 of this half
    v2f b;  b.x  = X[(size_t)kb*HW + pix];     b.y  = X[(size_t)(kb+1)*HW + pix];
    v2f at; at.x = tw[m*Cc + kb];              at.y = tw[m*Cc + kb + 1];
    v2f ap; ap.x = pw[m*Cc + kb];              ap.y = pw[m*Cc + kb + 1];
    v2f ag; ag.x = gw[m*Cc + kb];              ag.y = gw[m*Cc + kb + 1];
    ct = __builtin_amdgcn_wmma_f32_16x16x4_f32(false, at, false, b, (short)0, ct, false, false);
    cp = __builtin_amdgcn_wmma_f32_16x16x4_f32(false, ap, false, b, (short)0, cp, false, false);
    cg = __builtin_amdgcn_wmma_f32_16x16x4_f32(false, ag, false, b, (short)0, cg, false, false);
  }

  #pragma unroll
  for (int r = 0; r < 8; ++r) {
    int mm = r + 8*hf;                     // C/D: VGPR r holds rows r and r+8
    size_t o = ((size_t)n*CIc + mm)*HW + pix;
    thetaO[o] = ct[r] + tb[mm];
    phiO[o]   = cp[r] + pb[mm];
    gO[o]     = cg[r] + gb[mm];
  }
}

// ---------------------------------------------------------------------------
// Kernel 2: one workgroup (256 threads) per query.
//  - stage phi/g neighborhood (<=16x15x15) + query patch in LDS
//  - 81 window scores = 2*dot - |Cg|^2  (|Qp|^2 constant -> dropped)
//  - wave0: top-32 (lower-index tie-break, matches lax.top_k) + softmax
//  - weighted g-patch scattered into canvas with global atomics
// ---------------------------------------------------------------------------
__global__ __launch_bounds__(256) void attn_kernel(
    const float* __restrict__ theta, const float* __restrict__ phi,
    const float* __restrict__ gbuf, const int* __restrict__ flows,
    float* __restrict__ canvas)
{
  __shared__ float sPhi[CIc*15*16];
  __shared__ float sG  [CIc*15*16];
  __shared__ float sQ  [CIc*49];
  __shared__ float sDot[81];
  __shared__ float sCn [81];
  __shared__ int   sBdi[96];
  __shared__ int   sBdj[96];
  __shared__ float sWk [32];
  __shared__ int   sKdi[32];
  __shared__ int   sKdj[32];

  int q   = blockIdx.x;
  int n   = q / NQc;
  int qq  = q % NQc;
  int qci = (qq / nHc) * 4;
  int qcj = (qq % nHc) * 4;
  int tid = threadIdx.x;

  int f0 = flows[((size_t)n*2 + 0)*HW + qci*Wc + qcj];
  int f1 = flows[((size_t)n*2 + 1)*HW + qci*Wc + qcj];

  int cilo = clampi(qci + f0 - 4, 0, Hh-1);
  int cihi = clampi(qci + f0 + 4, 0, Hh-1);
  int cjlo = clampi(qcj + f1 - 4, 0, Wc-1);
  int cjhi = clampi(qcj + f1 + 4, 0, Wc-1);
  int row_lo = cilo - 3, col_lo = cjlo - 3;
  int nrows  = cihi + 3 - row_lo + 1;      // <= 15
  int ncols  = cjhi + 3 - col_lo + 1;      // <= 15

  // Stage phi & g neighborhood tiles (reflect padding applied on the fly).
  int tileE = CIc * nrows * ncols;
  for (int t = tid; t < tileE; t += 256) {
    int ch  = t / (nrows*ncols); int rem = t - ch*nrows*ncols;
    int di  = rem / ncols;       int dj  = rem - di*ncols;
    int rr  = reflect_idx(row_lo + di);
    int cc  = reflect_idx(col_lo + dj);
    size_t gi = ((size_t)n*CIc + ch)*HW + rr*Wc + cc;
    int li = ch*240 + di*16 + dj;          // dj stride padded to 16
    sPhi[li] = phi[gi];
    sG[li]   = gbuf[gi];
  }
  // Stage query patch Qp (theta patch at (qci+WH, qcj+WH)).
  for (int t = tid; t < CIc*49; t += 256) {
    int ch = t / 49; int pp = t - ch*49;
    int pi = pp / 7, pj = pp - pi*7;
    int rr = reflect_idx(qci - 3 + pi);
    int cc = reflect_idx(qcj - 3 + pj);
    sQ[t] = theta[((size_t)n*CIc + ch)*HW + rr*Wc + cc];
  }
  if (tid < 96) {
    if (tid < 81) {
      sDot[tid] = 0.f; sCn[tid] = 0.f;
      int oi = tid / 9 - 4, oj = tid % 9 - 4;
      int ci = clampi(qci + f0 + oi, 0, Hh-1);
      int cj = clampi(qcj + f1 + oj, 0, Wc-1);
      sBdi[tid] = ci - 3 - row_lo;
      sBdj[tid] = cj - 3 - col_lo;
    } else { sBdi[tid] = 0; sBdj[tid] = 0; }
  }
  __syncthreads();

  // dot(Qp, Cg_w) and |Cg_w|^2, split over (w, channel) tasks.
  for (int t = tid; t < 81*16; t += 256) {
    int w  = t >> 4, ch = t & 15;
    int base = ch*240 + sBdi[w]*16 + sBdj[w];
    const float* qp = &sQ[ch*49];
    float ds = 0.f, cs = 0.f;
    #pragma unroll
    for (int pi = 0; pi < 7; ++pi) {
      int b2 = base + pi*16;
      #pragma unroll
      for (int pj = 0; pj < 7; ++pj) {
        float pv = sPhi[b2 + pj];
        ds += qp[pi*7+pj] * pv;
        cs += pv * pv;
      }
    }
    atomicAdd(&sDot[w], ds);
    atomicAdd(&sCn[w],  cs);
  }
  __syncthreads();

  // Wave 0: top-32 selection + softmax.
  if (tid < 32) {
    float v0 = (tid      < 81) ? 2.f*sDot[tid]    - sCn[tid]    : -FLT_MAX;
    float v1 = (tid + 32 < 81) ? 2.f*sDot[tid+32] - sCn[tid+32] : -FLT_MAX;
    float v2 = (tid + 64 < 81) ? 2.f*sDot[tid+64] - sCn[tid+64] : -FLT_MAX;
    float myv = 0.f; int myi = 0;
    for (int k = 0; k < 32; ++k) {
      float bv = v0; int bi = tid;
      if (v1 > bv) { bv = v1; bi = tid + 32; }
      if (v2 > bv) { bv = v2; bi = tid + 64; }
      for (int off = 16; off >= 1; off >>= 1) {
        float ov = __shfl_xor(bv, off, 32);
        int   oi = __shfl_xor(bi, off, 32);
        if (ov > bv || (ov == bv && oi < bi)) { bv = ov; bi = oi; }
      }
      if (tid == k) { myv = bv; myi = bi; }
      if      (bi == tid)      v0 = -FLT_MAX;
      else if (bi == tid + 32) v1 = -FLT_MAX;
      else if (bi == tid + 64) v2 = -FLT_MAX;
    }
    float mx = __shfl(myv, 0, 32);
    float e  = expf(10.0f * (myv - mx));
    float s  = e;
    for (int off = 16; off >= 1; off >>= 1) s += __shfl_xor(s, off, 32);
    sWk[tid]  = e / s;
    sKdi[tid] = sBdi[myi];
    sKdj[tid] = sBdj[myi];
  }
  __syncthreads();

  // Weighted sum of the 32 selected g-patches, scatter-add to canvas.
  for (int e = tid; e < CIc*49; e += 256) {
    int ch = e / 49; int pp = e - ch*49;
    int pi = pp / 7, pj = pp - pi*7;
    float acc = 0.f;
    #pragma unroll
    for (int k = 0; k < 32; ++k)
      acc += sWk[k] * sG[ch*240 + (sKdi[k]+pi)*16 + (sKdj[k]+pj)];
    atomicAdd(&canvas[(((size_t)n*HCc + (qci+pi))*HCc + (qcj+pj))*CIc + ch], acc);
  }
}

// ---------------------------------------------------------------------------
// Kernel 3: agg = canvas/cnt, out = vid + w_w @ agg + w_b via WMMA.
// Per wave: 64(C)x16(pixel) output = 4 M-tiles x (K=16 in 4 wmma steps).
// ---------------------------------------------------------------------------
__global__ __launch_bounds__(256) void out_kernel(
    const float* __restrict__ vid, const float* __restrict__ canvas,
    const float* __restrict__ ww,  const float* __restrict__ wb,
    float* __restrict__ out)
{
  __shared__ float sAgg[8][16][17];
  int wave = threadIdx.x >> 5;
  int lane = threadIdx.x & 31;
  int gt   = blockIdx.x * 8 + wave;
  int n    = gt / (HW/16);
  int tile = gt % (HW/16);
  int col  = lane & 15;
  int hf   = lane >> 4;
  int pix  = tile*16 + col;
  int i = pix / Wc, j = pix % Wc;
  int ic = i + PPc, jc = j + PPc;

  // patch coverage counts (geometry only)
  int rci = 0, rcj = 0;
  for (int qv = 0; qv <= 92; qv += 4) {
    rci += (qv <= ic && ic <= qv + 6);
    rcj += (qv <= jc && jc <= qv + 6);
  }
  float invc = 1.0f / (float)(rci * rcj);

  const float* cb = canvas + (((size_t)n*HCc + ic)*HCc + jc)*CIc;
  for (int ch = hf*8; ch < hf*8 + 8; ++ch)
    sAgg[wave][col][ch] = cb[ch] * invc;
  __syncthreads();

  v8f acc0 = {0.f,0.f,0.f,0.f,0.f,0.f,0.f,0.f};
  v8f acc1 = {0.f,0.f,0.f,0.f,0.f,0.f,0.f,0.f};
  v8f acc2 = {0.f,0.f,0.f,0.f,0.f,0.f,0.f,0.f};
  v8f acc3 = {0.f,0.f,0.f,0.f,0.f,0.f,0.f,0.f};
  #pragma unroll
  for (int kk = 0; kk < 4; ++kk) {
    int kb = kk*4 + 2*hf;
    v2f b; b.x = sAgg[wave][col][kb]; b.y = sAgg[wave][col][kb+1];
    v2f a0; a0.x = ww[(0*16+col)*CIc + kb]; a0.y = ww[(0*16+col)*CIc + kb + 1];
    v2f a1; a1.x = ww[(1*16+col)*CIc + kb]; a1.y = ww[(1*16+col)*CIc + kb + 1];
    v2f a2; a2.x = ww[(2*16+col)*CIc + kb]; a2.y = ww[(2*16+col)*CIc + kb + 1];
    v2f a3; a3.x = ww[(3*16+col)*CIc + kb]; a3.y = ww[(3*16+col)*CIc + kb + 1];
    acc0 = __builtin_amdgcn_wmma_f32_16x16x4_f32(false, a0, false, b, (short)0, acc0, false, false);
    acc1 = __builtin_amdgcn_wmma_f32_16x16x4_f32(false, a1, false, b, (short)0, acc1, false, false);
    acc2 = __builtin_amdgcn_wmma_f32_16x16x4_f32(false, a2, false, b, (short)0, acc2, false, false);
    acc3 = __builtin_amdgcn_wmma_f32_16x16x4_f32(false, a3, false, b, (short)0, acc3, false, false);
  }

  #pragma unroll
  for (int r = 0; r < 8; ++r) {
    int mr = r + 8*hf;
    { int m = 0*16 + mr; size_t o = ((size_t)n*Cc + m)*HW + pix; out[o] = vid[o] + acc0[r] + wb[m]; }
    { int m = 1*16 + mr; size_t o = ((size_t)n*Cc + m)*HW + pix; out[o] = vid[o] + acc1[r] + wb[m]; }
    { int m = 2*16 + mr; size_t o = ((size_t)n*Cc + m)*HW + pix; out[o] = vid[o] + acc2[r] + wb[m]; }
    { int m = 3*16 + mr; size_t o = ((size_t)n*Cc + m)*HW + pix; out[o] = vid[o] + acc3[r] + wb[m]; }
  }
}

// ---------------------------------------------------------------------------
extern "C" void kernel_launch(void* const* d_in, const int* in_sizes, int n_in,
                              void* d_out, int out_size, void* d_ws, size_t ws_size,
                              hipStream_t stream) {
  const float* vid   = (const float*)d_in[0];
  const int*   flows = (const int*)  d_in[1];
  const float* tw    = (const float*)d_in[2];
  const float* tb    = (const float*)d_in[3];
  const float* pw    = (const float*)d_in[4];
  const float* pb    = (const float*)d_in[5];
  const float* gw    = (const float*)d_in[6];
  const float* gb    = (const float*)d_in[7];
  const float* wwp   = (const float*)d_in[8];
  const float* wbp   = (const float*)d_in[9];
  float* out = (float*)d_out;

  size_t projElems = (size_t)NB * CIc * HW;        // 589824 floats each
  float* thetaW = (float*)d_ws;
  float* phiW   = thetaW + projElems;
  float* gW     = phiW   + projElems;
  float* canvas = gW     + projElems;
  size_t canvasElems = (size_t)NB * HCc * HCc * CIc; // 665856 floats

  hipMemsetAsync(canvas, 0, canvasElems * sizeof(float), stream);
  proj_kernel<<<288, 256, 0, stream>>>(vid, tw, tb, pw, pb, gw, gb, thetaW, phiW, gW);
  attn_kernel<<<NB*NQc, 256, 0, stream>>>(thetaW, phiW, gW, flows, canvas);
  out_kernel<<<288, 256, 0, stream>>>(vid, canvas, wwp, wbp, out);
}